// ComplexBatchNormalization_57784490000847
// MI455X (gfx1250) — compile-verified
//
#include <hip/hip_runtime.h>
#include <math.h>

// ---------------- problem constants (from reference setup_inputs) ----------
constexpr int   C              = 256;                 // channels (innermost dim)
constexpr int   NROWS          = 32 * 56 * 56;        // 100352 samples per channel
constexpr int   NB1            = 392;                 // reduction blocks
constexpr int   ROWS_PER_BLOCK = NROWS / NB1;         // 256
constexpr int   TROWS          = 16;                  // rows per TDM tile
constexpr int   TILES          = ROWS_PER_BLOCK / TROWS; // 16
constexpr unsigned TILE_FLOATS = (unsigned)TROWS * C;        // 4096
constexpr unsigned TILE_BYTES  = TILE_FLOATS * 4u;           // 16 KB
#define EPSV 1e-3f

// ---------------- CDNA5 Tensor Data Mover descriptor + issue ---------------
typedef __attribute__((ext_vector_type(4))) unsigned int u32x4;
typedef __attribute__((ext_vector_type(8))) int          i32x8;
typedef __attribute__((ext_vector_type(4))) int          i32x4;

// Loads a [TROWS x 256] f32 tile (row-major, stride 256 floats) from global
// memory into LDS at byte offset lds_off.  Issued per-wave (EXEC ignored),
// tracked by TENSORcnt.
__device__ __forceinline__ void tdm_load_tile(const float* gsrc, unsigned lds_off) {
  const unsigned long long ga = (unsigned long long)(size_t)gsrc;
  u32x4 g0;
  g0.x = 1u;                                                // count=1, user descriptor
  g0.y = lds_off;                                           // lds_addr (bytes)
  g0.z = (unsigned)(ga & 0xffffffffull);                    // global_addr[31:0]
  g0.w = (unsigned)((ga >> 32) & 0x01ffffffull) | (2u << 30); // addr[56:32] | type=2
  i32x8 g1;
  g1[0] = (int)(2u << 16);                 // workgroup_mask=0, data_size=2 (4B)
  g1[1] = (int)((unsigned)C << 16);        // tensor_dim0 = 256   (bits 79:48, low 16)
  g1[2] = (int)((unsigned)TROWS << 16);    // tensor_dim1 = TROWS (bits 111:80, low 16)
  g1[3] = (int)((unsigned)C << 16);        // tensor_dim1 hi = 0, tile_dim0 = 256
  g1[4] = (int)TROWS;                      // tile_dim1 = TROWS, tile_dim2 = 0
  g1[5] = C;                               // tensor_dim0_stride = 256 (low 32)
  g1[6] = 0;                               // stride0 hi = 0, stride1 lo = 0
  g1[7] = 0;                               // stride1 hi = 0
  const i32x4 zz4 = {0, 0, 0, 0};
  const i32x8 zz8 = {0, 0, 0, 0, 0, 0, 0, 0};
  __builtin_amdgcn_tensor_load_to_lds(g0, g1, zz4, zz4, zz8, 0);
}

// ---------------- pass 1: per-block partial moments ------------------------
// LDS layout (dynamic): [buf0 real 16K][buf0 imag 16K][buf1 real 16K][buf1 imag 16K]
__global__ void __launch_bounds__(256)
cbn_partial_kernel(const float* __restrict__ xr, const float* __restrict__ xi,
                   float* __restrict__ partials) {
  extern __shared__ float smem[];
  const int c = threadIdx.x;                       // thread == channel
  const long long row0 = (long long)blockIdx.x * ROWS_PER_BLOCK;

  if (threadIdx.x == 0) {                          // wave 0 issues DMA (EXEC ignored)
    tdm_load_tile(xr + row0 * C, 0u);
    tdm_load_tile(xi + row0 * C, TILE_BYTES);
  }

  float sr = 0.f, si = 0.f, srr = 0.f, sri = 0.f, sii = 0.f;
  for (int tl = 0; tl < TILES; ++tl) {
    if (threadIdx.x == 0) {
      if (tl + 1 < TILES) {                        // prefetch next tile, keep 2 in flight
        const unsigned boff = ((tl + 1) & 1) ? 2u * TILE_BYTES : 0u;
        const long long trow = row0 + (long long)(tl + 1) * TROWS;
        tdm_load_tile(xr + trow * C, boff);
        tdm_load_tile(xi + trow * C, boff + TILE_BYTES);
        __builtin_amdgcn_s_wait_tensorcnt(2);      // current tile's 2 ops done
      } else {
        __builtin_amdgcn_s_wait_tensorcnt(0);
      }
    }
    __syncthreads();                               // publish tile to all 8 waves

    const float* bR = smem + ((tl & 1) ? 2u * TILE_FLOATS : 0u);
    const float* bI = bR + TILE_FLOATS;
#pragma unroll
    for (int r = 0; r < TROWS; ++r) {
      const float a = bR[r * C + c];
      const float b = bI[r * C + c];
      sr += a; si += b;
      srr = __builtin_fmaf(a, a, srr);
      sri = __builtin_fmaf(a, b, sri);
      sii = __builtin_fmaf(b, b, sii);
    }
    __syncthreads();                               // buffer free for reuse
  }

  const int base = blockIdx.x * C + c;             // partials[5][NB1][C]
  partials[0 * NB1 * C + base] = sr;
  partials[1 * NB1 * C + base] = si;
  partials[2 * NB1 * C + base] = srr;
  partials[3 * NB1 * C + base] = sri;
  partials[4 * NB1 * C + base] = sii;
}

// ---------------- pass 2: finalize stats, fold gamma/beta ------------------
__global__ void __launch_bounds__(256)
cbn_stats_kernel(const float* __restrict__ partials,
                 const float* __restrict__ gr, const float* __restrict__ gi,
                 const float* __restrict__ br, const float* __restrict__ bi,
                 float* __restrict__ coef) {
  const int c = threadIdx.x;
  float sr = 0.f, si = 0.f, srr = 0.f, sri = 0.f, sii = 0.f;
  for (int bk = 0; bk < NB1; ++bk) {
    const int o = bk * C + c;
    sr  += partials[0 * NB1 * C + o];
    si  += partials[1 * NB1 * C + o];
    srr += partials[2 * NB1 * C + o];
    sri += partials[3 * NB1 * C + o];
    sii += partials[4 * NB1 * C + o];
  }
  const float invN = 1.0f / (float)NROWS;
  const float mr = sr * invN, mi = si * invN;
  const float vrr = srr * invN - mr * mr + EPSV;
  const float vri = sri * invN - mr * mi;
  const float vii = sii * invN - mi * mi + EPSV;

  const float det = vrr * vii - vri * vri;         // A = inv(V)
  const float a =  vii / det;
  const float b = -vri / det;
  const float d =  vrr / det;
  const float s = sqrtf(a * d - b * b);            // S = sqrtm(A)
  const float t = sqrtf(a + d + 2.0f * s);
  const float s00 = (a + s) / t;
  const float s01 = b / t;
  const float s11 = (d + s) / t;

  const float Gr = gr[c], Gi = gi[c];
  const float Arr = Gr * s00 - Gi * s01;           // fold complex gamma into S
  const float Ari = Gr * s01 - Gi * s11;
  const float Air = Gr * s01 + Gi * s00;
  const float Aii = Gr * s11 + Gi * s01;
  coef[0 * C + c] = Arr;
  coef[1 * C + c] = Ari;
  coef[2 * C + c] = Air;
  coef[3 * C + c] = Aii;
  coef[4 * C + c] = br[c] - Arr * mr - Ari * mi;   // fold mean + beta
  coef[5 * C + c] = bi[c] - Air * mr - Aii * mi;
}

// ---------------- pass 3: elementwise apply (streaming, 128-bit) -----------
__global__ void __launch_bounds__(256)
cbn_apply_kernel(const float* __restrict__ xr, const float* __restrict__ xi,
                 const float* __restrict__ coef, float* __restrict__ out) {
  const int g  = blockIdx.x * 256 + threadIdx.x;
  const int cg = g & 63;                           // channel group (4 channels)
  const int c4 = cg * 4;
  const int rowStride = (gridDim.x * 256) >> 6;

  const float4 Arr = *(const float4*)(coef + 0 * C + c4);
  const float4 Ari = *(const float4*)(coef + 1 * C + c4);
  const float4 Air = *(const float4*)(coef + 2 * C + c4);
  const float4 Aii = *(const float4*)(coef + 3 * C + c4);
  const float4 Cr  = *(const float4*)(coef + 4 * C + c4);
  const float4 Ci  = *(const float4*)(coef + 5 * C + c4);

  for (int row = g >> 6; row < NROWS; row += rowStride) {
    const int ib = row * C + c4;
    const float4 vr = *(const float4*)(xr + ib);
    const float4 vi = *(const float4*)(xi + ib);
    float4 o0, o1;                                 // interleaved (re,im) pairs
    o0.x = Arr.x * vr.x + Ari.x * vi.x + Cr.x;
    o0.y = Air.x * vr.x + Aii.x * vi.x + Ci.x;
    o0.z = Arr.y * vr.y + Ari.y * vi.y + Cr.y;
    o0.w = Air.y * vr.y + Aii.y * vi.y + Ci.y;
    o1.x = Arr.z * vr.z + Ari.z * vi.z + Cr.z;
    o1.y = Air.z * vr.z + Aii.z * vi.z + Ci.z;
    o1.z = Arr.w * vr.w + Ari.w * vi.w + Cr.w;
    o1.w = Air.w * vr.w + Aii.w * vi.w + Ci.w;
    float* op = out + (size_t)row * (2 * C) + 2 * c4;
    *(float4*)(op)     = o0;
    *(float4*)(op + 4) = o1;
  }
}

// ---------------- host entry ----------------------------------------------
extern "C" void kernel_launch(void* const* d_in, const int* in_sizes, int n_in,
                              void* d_out, int out_size, void* d_ws, size_t ws_size,
                              hipStream_t stream) {
  (void)in_sizes; (void)n_in; (void)out_size; (void)ws_size;
  const float* xr = (const float*)d_in[0];
  const float* xi = (const float*)d_in[1];
  const float* gr = (const float*)d_in[2];
  const float* gi = (const float*)d_in[3];
  const float* br = (const float*)d_in[4];
  const float* bi = (const float*)d_in[5];
  float* out      = (float*)d_out;

  float* partials = (float*)d_ws;                  // 5*NB1*C floats (~2 MB)
  float* coef     = partials + 5 * NB1 * C;        // 6*C floats

  cbn_partial_kernel<<<NB1, 256, 4 * TILE_BYTES, stream>>>(xr, xi, partials);
  cbn_stats_kernel<<<1, 256, 0, stream>>>(partials, gr, gi, br, bi, coef);
  cbn_apply_kernel<<<1024, 256, 0, stream>>>(xr, xi, coef, out);
}